// PointNetSetAbstraction_37847251812419
// MI455X (gfx1250) — compile-verified
//
#include <hip/hip_runtime.h>
#include <hip/hip_bf16.h>
#include <math.h>

typedef __attribute__((ext_vector_type(2))) float v2f;
typedef __attribute__((ext_vector_type(8))) float v8f;

#define NPTS   4096
#define NBATCH 16
#define NPOINT 1024
#define NSAMP  32
#define MTOT   524288   /* 16*1024*32 = 2^19 */

// ---------------------------------------------------------------------------
// 1) Farthest point sampling: one block per batch, xyz staged in LDS.
//    Emits new_xyz directly into d_out (first 16*1024*3 floats).
// ---------------------------------------------------------------------------
__global__ __launch_bounds__(1024) void fps_kernel(const float* __restrict__ xyz,
                                                   float* __restrict__ new_xyz) {
    __shared__ float sx[NPTS], sy[NPTS], sz[NPTS];
    __shared__ float rv[1024];
    __shared__ int   ri[1024];
    __shared__ int   sfar;
    const int b = blockIdx.x, t = threadIdx.x;
    const float* bx = xyz + (size_t)b * NPTS * 3;
    for (int i = t; i < NPTS; i += 1024) {
        sx[i] = bx[i * 3 + 0];
        sy[i] = bx[i * 3 + 1];
        sz[i] = bx[i * 3 + 2];
    }
    if (t == 0) sfar = 0;
    __syncthreads();

    const int p0 = t * 4, p1 = p0 + 1, p2 = p0 + 2, p3 = p0 + 3;
    float d0 = 1e10f, d1 = 1e10f, d2 = 1e10f, d3 = 1e10f;

    for (int it = 0; it < NPOINT; ++it) {
        const int far = sfar;
        const float cx = sx[far], cy = sy[far], cz = sz[far];
        if (t == 0) {
            float* o = new_xyz + ((size_t)b * NPOINT + it) * 3;
            o[0] = cx; o[1] = cy; o[2] = cz;
        }
        float e;
        e = (sx[p0]-cx)*(sx[p0]-cx) + (sy[p0]-cy)*(sy[p0]-cy) + (sz[p0]-cz)*(sz[p0]-cz); d0 = fminf(d0, e);
        e = (sx[p1]-cx)*(sx[p1]-cx) + (sy[p1]-cy)*(sy[p1]-cy) + (sz[p1]-cz)*(sz[p1]-cz); d1 = fminf(d1, e);
        e = (sx[p2]-cx)*(sx[p2]-cx) + (sy[p2]-cy)*(sy[p2]-cy) + (sz[p2]-cz)*(sz[p2]-cz); d2 = fminf(d2, e);
        e = (sx[p3]-cx)*(sx[p3]-cx) + (sy[p3]-cy)*(sy[p3]-cy) + (sz[p3]-cz)*(sz[p3]-cz); d3 = fminf(d3, e);

        float bv = d0; int bi = p0;                 // ties -> lowest index (argmax semantics)
        if (d1 > bv) { bv = d1; bi = p1; }
        if (d2 > bv) { bv = d2; bi = p2; }
        if (d3 > bv) { bv = d3; bi = p3; }
        rv[t] = bv; ri[t] = bi;
        __syncthreads();
        for (int off = 512; off > 0; off >>= 1) {
            if (t < off) {
                const float ov = rv[t + off]; const int oi = ri[t + off];
                if (ov > rv[t] || (ov == rv[t] && oi < ri[t])) { rv[t] = ov; ri[t] = oi; }
            }
            __syncthreads();
        }
        if (t == 0) sfar = ri[0];
        __syncthreads();
    }
}

// ---------------------------------------------------------------------------
// 2) Ball query + gather + concat into channel-major X0[c][m], m contiguous.
//    Channel 67 is the zero pad row for the K=68 WMMA loop of layer 0.
// ---------------------------------------------------------------------------
__global__ __launch_bounds__(1024) void group_kernel(const float* __restrict__ xyz,
                                                     const float* __restrict__ points,
                                                     const float* __restrict__ new_xyz,
                                                     float* __restrict__ X0) {
    __shared__ float sx[NPTS], sy[NPTS], sz[NPTS];
    const int b = blockIdx.x, s = threadIdx.x;
    const float* bx = xyz + (size_t)b * NPTS * 3;
    for (int i = s; i < NPTS; i += 1024) {
        sx[i] = bx[i * 3 + 0];
        sy[i] = bx[i * 3 + 1];
        sz[i] = bx[i * 3 + 2];
    }
    __syncthreads();

    const float* c = new_xyz + ((size_t)b * NPOINT + s) * 3;
    const float cx = c[0], cy = c[1], cz = c[2];
    const float cn = cx * cx + cy * cy + cz * cz;

    int gi[NSAMP];
    int cnt = 0;
    for (int p = 0; p < NPTS && cnt < NSAMP; ++p) {
        const float px = sx[p], py = sy[p], pz = sz[p];
        const float dd = cn + px * px + py * py + pz * pz - 2.0f * (cx * px + cy * py + cz * pz);
        const float dist = sqrtf(fmaxf(dd, 0.0f));
        if (!(dist > 0.81f)) gi[cnt++] = p;   // reference compares sqrt(d2) > radius^2
    }
    const int g0 = (cnt > 0) ? gi[0] : 0;
    for (int k = cnt; k < NSAMP; ++k) gi[k] = g0;

    const size_t mb = ((size_t)b * NPOINT + s) * NSAMP;
    const float* bp = points + (size_t)b * NPTS * 64;
    for (int k = 0; k < NSAMP; ++k) {
        const int g = gi[k];
        X0[0 * (size_t)MTOT + mb + k] = sx[g] - cx;
        X0[1 * (size_t)MTOT + mb + k] = sy[g] - cy;
        X0[2 * (size_t)MTOT + mb + k] = sz[g] - cz;
        const float* pr = bp + (size_t)g * 64;
        for (int ci = 0; ci < 64; ++ci)
            X0[(size_t)(3 + ci) * MTOT + mb + k] = pr[ci];
        X0[(size_t)67 * MTOT + mb + k] = 0.0f;   // zero pad channel
    }
}

// ---------------------------------------------------------------------------
// 3) GEMM via V_WMMA_F32_16X16X4_F32 with output-channel blocking:
//    each wave owns one 16-wide m-tile and ALL NT=O/16 o-tiles, so every
//    B-fragment (the big X operand) is loaded from HBM exactly once and
//    reused by NT WMMAs. Weights are zero-padded to CPAD in LDS.
//    A-frag (16x4 f32): lanes 0-15 = rows, lane>>4 selects K pair.
//    B-frag (4x16 f32): lanes 0-15 = cols -> coalesced 64B half-wave loads.
//    C/D: VGPR j, lanes 0-15 -> M=j, lanes 16-31 -> M=8+j.
// ---------------------------------------------------------------------------
template <int NT, int CPAD>
__global__ __launch_bounds__(256) void gemm_wmma(const float* __restrict__ W,
                                                 const float* __restrict__ bias,
                                                 const float* __restrict__ X,
                                                 float* __restrict__ Y,
                                                 int Cact) {
    __shared__ float sW[NT * 16 * CPAD];             // <= 128x64 = 8192 floats (32KB)
    const int tid = threadIdx.x;
    const int total = NT * 16 * CPAD;
    for (int i = tid; i < total; i += 256) {
        const int o = i / CPAD, cc = i % CPAD;
        sW[i] = (cc < Cact) ? W[o * Cact + cc] : 0.0f;
    }
    __syncthreads();

    const int lane   = tid & 31;
    const int wave   = tid >> 5;
    const int m_base = (blockIdx.x * 8 + wave) * 16; // wave owns one m-tile
    const int khalf  = (lane >> 4) * 2;
    const int col_b  = m_base + (lane & 15);
    const int m_row  = (lane & 16) ? 8 : 0;
    const int arow   = lane & 15;

    v8f acc[NT];
#pragma unroll
    for (int t = 0; t < NT; ++t)
#pragma unroll
        for (int j = 0; j < 8; ++j) acc[t][j] = bias[t * 16 + m_row + j];

#pragma unroll
    for (int k = 0; k < CPAD; k += 4) {
        v2f bfr;
        bfr.x = X[(size_t)(k + khalf) * MTOT + col_b];
        bfr.y = X[(size_t)(k + khalf + 1) * MTOT + col_b];
#pragma unroll
        for (int t = 0; t < NT; ++t) {
            v2f a;
            a.x = sW[(t * 16 + arow) * CPAD + k + khalf];
            a.y = sW[(t * 16 + arow) * CPAD + k + khalf + 1];
            acc[t] = __builtin_amdgcn_wmma_f32_16x16x4_f32(false, a, false, bfr,
                                                           (short)0, acc[t], false, false);
        }
    }

#pragma unroll
    for (int t = 0; t < NT; ++t)
#pragma unroll
        for (int j = 0; j < 8; ++j)
            Y[(size_t)(t * 16 + m_row + j) * MTOT + col_b] = acc[t][j];
}

// ---------------------------------------------------------------------------
// 4) Deterministic two-stage per-channel reduction for batch-norm stats.
// ---------------------------------------------------------------------------
__global__ __launch_bounds__(256) void chan_reduce(const float* __restrict__ Y,
                                                   float* __restrict__ psum,
                                                   float* __restrict__ psq) {
    const int ch = blockIdx.y, chunk = blockIdx.x;
    const int CH = MTOT / 64;                        // 8192 per chunk
    const size_t base = (size_t)ch * MTOT + (size_t)chunk * CH;
    float s = 0.f, q = 0.f;
    for (int i = threadIdx.x; i < CH; i += 256) {
        const float v = Y[base + i];
        s += v; q += v * v;
    }
    __shared__ float ss[256], sq[256];
    ss[threadIdx.x] = s; sq[threadIdx.x] = q;
    __syncthreads();
    for (int off = 128; off > 0; off >>= 1) {
        if (threadIdx.x < off) {
            ss[threadIdx.x] += ss[threadIdx.x + off];
            sq[threadIdx.x] += sq[threadIdx.x + off];
        }
        __syncthreads();
    }
    if (threadIdx.x == 0) {
        psum[ch * 64 + chunk] = ss[0];
        psq[ch * 64 + chunk]  = sq[0];
    }
}

__global__ void bn_finalize(const float* __restrict__ psum, const float* __restrict__ psq,
                            float* __restrict__ stats, int O) {
    const int ch = threadIdx.x;
    if (ch < O) {
        float s = 0.f, q = 0.f;
        for (int i = 0; i < 64; ++i) { s += psum[ch * 64 + i]; q += psq[ch * 64 + i]; }
        const float m   = s / (float)MTOT;
        const float var = q / (float)MTOT - m * m;
        stats[ch]       = m;
        stats[128 + ch] = 1.0f / sqrtf(var + 1e-5f);
    }
}

__global__ __launch_bounds__(256) void bn_relu(float* __restrict__ Y,
                                               const float* __restrict__ stats,
                                               const float* __restrict__ g,
                                               const float* __restrict__ be, int O) {
    const size_t total = (size_t)O * MTOT;
    for (size_t i = (size_t)blockIdx.x * 256 + threadIdx.x; i < total;
         i += (size_t)gridDim.x * 256) {
        const int ch = (int)(i >> 19);               // MTOT = 2^19
        const float v = (Y[i] - stats[ch]) * stats[128 + ch] * g[ch] + be[ch];
        Y[i] = v > 0.f ? v : 0.f;
    }
}

// ---------------------------------------------------------------------------
// 5) Max over K=32 + transpose to (B, S, 128) into d_out after new_xyz.
// ---------------------------------------------------------------------------
__global__ __launch_bounds__(256) void maxout_kernel(const float* __restrict__ Y2,
                                                     float* __restrict__ outp) {
    const int i  = blockIdx.x * 256 + threadIdx.x;   // < 16*1024*128 exactly
    const int o  = i & 127;
    const int bs = i >> 7;                           // b*1024 + s
    const size_t base = (size_t)o * MTOT + (size_t)bs * NSAMP;
    float m = Y2[base];
    for (int k = 1; k < NSAMP; ++k) m = fmaxf(m, Y2[base + k]);
    outp[(size_t)bs * 128 + o] = m;
}

// ---------------------------------------------------------------------------
extern "C" void kernel_launch(void* const* d_in, const int* in_sizes, int n_in,
                              void* d_out, int out_size, void* d_ws, size_t ws_size,
                              hipStream_t stream) {
    const float* xyz    = (const float*)d_in[0];
    const float* points = (const float*)d_in[1];
    const float* w0 = (const float*)d_in[2];
    const float* b0 = (const float*)d_in[3];
    const float* g0 = (const float*)d_in[4];
    const float* be0= (const float*)d_in[5];
    const float* w1 = (const float*)d_in[6];
    const float* b1 = (const float*)d_in[7];
    const float* g1 = (const float*)d_in[8];
    const float* be1= (const float*)d_in[9];
    const float* w2 = (const float*)d_in[10];
    const float* b2 = (const float*)d_in[11];
    const float* g2 = (const float*)d_in[12];
    const float* be2= (const float*)d_in[13];
    float* out = (float*)d_out;

    char* ws = (char*)d_ws;
    float* R0 = (float*)ws;                                   // 68 x M  (X0, later Y1)
    float* R2 = (float*)(ws + (size_t)68 * MTOT * 4);         // 128 x M (Y0, later Y2)
    float* psum  = (float*)(ws + (size_t)(68 + 128) * MTOT * 4);
    float* psq   = psum + 128 * 64;
    float* stats = psq  + 128 * 64;

    const int gemm_blocks = (MTOT / 16) / 8;                  // one wave per m-tile

    fps_kernel  <<<NBATCH, 1024, 0, stream>>>(xyz, out);
    group_kernel<<<NBATCH, 1024, 0, stream>>>(xyz, points, out, R0);

    // layer 0: 67(->68) -> 64
    gemm_wmma<4, 68><<<gemm_blocks, 256, 0, stream>>>(w0, b0, R0, R2, 67);
    chan_reduce<<<dim3(64, 64), 256, 0, stream>>>(R2, psum, psq);
    bn_finalize<<<1, 128, 0, stream>>>(psum, psq, stats, 64);
    bn_relu    <<<4096, 256, 0, stream>>>(R2, stats, g0, be0, 64);

    // layer 1: 64 -> 64
    gemm_wmma<4, 64><<<gemm_blocks, 256, 0, stream>>>(w1, b1, R2, R0, 64);
    chan_reduce<<<dim3(64, 64), 256, 0, stream>>>(R0, psum, psq);
    bn_finalize<<<1, 128, 0, stream>>>(psum, psq, stats, 64);
    bn_relu    <<<4096, 256, 0, stream>>>(R0, stats, g1, be1, 64);

    // layer 2: 64 -> 128
    gemm_wmma<8, 64><<<gemm_blocks, 256, 0, stream>>>(w2, b2, R0, R2, 64);
    chan_reduce<<<dim3(64, 128), 256, 0, stream>>>(R2, psum, psq);
    bn_finalize<<<1, 128, 0, stream>>>(psum, psq, stats, 128);
    bn_relu    <<<8192, 256, 0, stream>>>(R2, stats, g2, be2, 128);

    // max over K + transpose; new_points starts after new_xyz (16*1024*3 floats)
    maxout_kernel<<<(NBATCH * NPOINT * 128) / 256, 256, 0, stream>>>(R2, out + 49152);
}